// CumulativeNormalizer_43052752175396
// MI455X (gfx1250) — compile-verified
//
#include <hip/hip_runtime.h>

// ---------------------------------------------------------------------------
// Cumulative (causal) normalization along T for x[B,C,T] fp32.
// out[t] = (x[t] - mean(x[0..t])) / sqrt(var(x[0..t]) + EPS)
//
// Roofline: 268 MB min traffic -> ~11.5 us @ 23.3 TB/s; ~13 VALU ops/element
// after amortizing the scan over 8 elements/thread keeps us bandwidth-bound.
// Single-pass hierarchical (sum, sumsq) scan; CDNA5 async global->LDS double
// buffering (ASYNCcnt); wave32 shuffle scan; NT vector stores.
// ---------------------------------------------------------------------------

#define EPS_F   1e-4f
#define T_LEN   8192
#define BLOCK   256
#define NWAVES  (BLOCK / 32)
#define VEC     8
#define CHUNK   (BLOCK * VEC)      // 2048 floats per chunk
#define NCHUNK  (T_LEN / CHUNK)    // 4 chunks per row

typedef float f32x4 __attribute__((ext_vector_type(4)));
typedef int   v4i   __attribute__((vector_size(16)));   // matches builtin param

// ---- CDNA5 async global->LDS path (ASYNCcnt), with safe fallback ----------
#if defined(__has_builtin)
#  if __has_builtin(__builtin_amdgcn_global_load_async_to_lds_b128)
#    define HAVE_ASYNC_LDS 1
#  endif
#  if __has_builtin(__builtin_amdgcn_s_wait_asynccnt)
#    define HAVE_WAIT_ASYNC 1
#  endif
#endif

// Copy 32B per lane: two b128 async ops sharing base pointers, using the
// instruction immediate offset (ISA: INST_OFFSET is added to BOTH the global
// and LDS addresses for async loads).
__device__ __forceinline__ void async_copy32(const float* gsrc, float* ldst) {
#if defined(HAVE_ASYNC_LDS)
  __builtin_amdgcn_global_load_async_to_lds_b128(
      (__attribute__((address_space(1))) v4i*)gsrc,
      (__attribute__((address_space(3))) v4i*)ldst, /*offset*/ 0,  /*cpol*/ 0);
  __builtin_amdgcn_global_load_async_to_lds_b128(
      (__attribute__((address_space(1))) v4i*)gsrc,
      (__attribute__((address_space(3))) v4i*)ldst, /*offset*/ 16, /*cpol*/ 0);
#else
  ((f32x4*)ldst)[0] = ((const f32x4*)gsrc)[0];
  ((f32x4*)ldst)[1] = ((const f32x4*)gsrc)[1];
#endif
}

#if defined(HAVE_ASYNC_LDS)
#  if defined(HAVE_WAIT_ASYNC)
#    define WAIT_ASYNC(N)                                     \
       do { __builtin_amdgcn_s_wait_asynccnt(N);              \
            __asm__ __volatile__("" ::: "memory"); } while (0)
#  else
#    define WAIT_ASYNC(N)                                     \
       __asm__ __volatile__("s_wait_asynccnt %0" :: "i"(N) : "memory")
#  endif
#else
#  define WAIT_ASYNC(N) do { } while (0)
#endif

#if defined(__has_builtin) && __has_builtin(__builtin_amdgcn_rcpf) && __has_builtin(__builtin_amdgcn_rsqf)
#  define RCPF(v)  __builtin_amdgcn_rcpf(v)
#  define RSQF(v)  __builtin_amdgcn_rsqf(v)
#else
#  define RCPF(v)  (1.0f / (v))
#  define RSQF(v)  rsqrtf(v)
#endif

// ---------------------------------------------------------------------------

__global__ __launch_bounds__(BLOCK)
void CumulativeNormalizer_43052752175396_kernel(const float* __restrict__ x,
                                                float* __restrict__ out) {
  __shared__ __align__(16) float  s_data[2][CHUNK];   // async double buffer (16 KB)
  __shared__ float2               s_wsum[NWAVES];     // per-wave (sum, sumsq)

  const int    tid  = threadIdx.x;
  const int    lane = tid & 31;
  const int    wid  = tid >> 5;
  const size_t row  = blockIdx.x;

  const float* xrow = x   + row * (size_t)T_LEN;
  float*       orow = out + row * (size_t)T_LEN;

  // Running carry of (sum, sumsq) over all previous chunks (uniform per block).
  float carry_s = 0.0f, carry_q = 0.0f;

  // Prologue: start streaming chunk 0 into buffer 0 (2 async ops per lane).
  async_copy32(xrow + tid * VEC, &s_data[0][tid * VEC]);

#pragma unroll
  for (int c = 0; c < NCHUNK; ++c) {
    const int cur = c & 1;

    // Prefetch next chunk while processing the current one, then wait until
    // only the prefetch's 2 async ops are outstanding (current chunk landed).
    if (c + 1 < NCHUNK) {
      async_copy32(xrow + (size_t)(c + 1) * CHUNK + tid * VEC,
                   &s_data[cur ^ 1][tid * VEC]);
      WAIT_ASYNC(2);
    } else {
      WAIT_ASYNC(0);
    }

    // Each lane reads back exactly the 32B it asynchronously loaded.
    const f32x4 xa = ((const f32x4*)&s_data[cur][tid * VEC])[0];
    const f32x4 xb = ((const f32x4*)&s_data[cur][tid * VEC])[1];
    const float e[VEC] = {xa.x, xa.y, xa.z, xa.w, xb.x, xb.y, xb.z, xb.w};

    // Per-thread partial (sum, sumsq) over its 8 elements.
    float ts = 0.0f, tq = 0.0f;
#pragma unroll
    for (int i = 0; i < VEC; ++i) { ts += e[i]; tq += e[i] * e[i]; }

    // Inclusive wave32 scan of thread partials (ds_bpermute shuffles).
    float is = ts, iq = tq;
#pragma unroll
    for (int d = 1; d < 32; d <<= 1) {
      const float us = __shfl_up(is, (unsigned)d, 32);
      const float uq = __shfl_up(iq, (unsigned)d, 32);
      if (lane >= d) { is += us; iq += uq; }
    }

    // Publish wave totals, combine across the 8 waves (broadcast LDS reads).
    if (lane == 31) s_wsum[wid] = make_float2(is, iq);
    __syncthreads();
    float woff_s = 0.0f, woff_q = 0.0f, tot_s = 0.0f, tot_q = 0.0f;
#pragma unroll
    for (int j = 0; j < NWAVES; ++j) {
      const float2 w = s_wsum[j];
      if (j < wid) { woff_s += w.x; woff_q += w.y; }
      tot_s += w.x; tot_q += w.y;
    }
    __syncthreads();  // protect s_wsum for the next chunk

    // Exclusive prefix for this thread's first element.
    float run_s = carry_s + woff_s + (is - ts);
    float run_q = carry_q + woff_q + (iq - tq);
    carry_s += tot_s;
    carry_q += tot_q;

    const int tbase = c * CHUNK + tid * VEC;  // global time index of e[0]

    float o[VEC];
#pragma unroll
    for (int i = 0; i < VEC; ++i) {
      run_s += e[i];
      run_q += e[i] * e[i];
      const float inv  = RCPF((float)(tbase + i + 1));
      const float mean = run_s * inv;
      const float var  = run_q * inv - mean * mean;
      o[i] = (e[i] - mean) * RSQF(var + EPS_F);
    }

    // Write-once output: two non-temporal 16B stores.
    const f32x4 o0 = {o[0], o[1], o[2], o[3]};
    const f32x4 o1 = {o[4], o[5], o[6], o[7]};
    __builtin_nontemporal_store(o0, (f32x4*)(orow + tbase));
    __builtin_nontemporal_store(o1, (f32x4*)(orow + tbase + 4));
  }
}

// ---------------------------------------------------------------------------

extern "C" void kernel_launch(void* const* d_in, const int* in_sizes, int n_in,
                              void* d_out, int out_size, void* d_ws, size_t ws_size,
                              hipStream_t stream) {
  const float* x   = (const float*)d_in[0];
  float*       out = (float*)d_out;
  const int    rows = in_sizes[0] / T_LEN;  // 16 * 256 = 4096 independent rows

  CumulativeNormalizer_43052752175396_kernel<<<dim3(rows), dim3(BLOCK), 0, stream>>>(x, out);
}